// TrajGRUCell_31353261261051
// MI455X (gfx1250) — compile-verified
//
#include <hip/hip_runtime.h>
#include <hip/hip_bf16.h>

typedef __attribute__((ext_vector_type(2))) float v2f;
typedef __attribute__((ext_vector_type(4))) float f4;
typedef __attribute__((ext_vector_type(8))) float v8f;

#define BATCH 16
#define HH 128
#define WW 128
#define HWSZ (HH * WW)
#define CIN 32
#define HID 64
#define C3 192
#define CCCH 16   // 2*CONNECTION
#define NW 4

// order-preserving float<->uint map for atomic min/max
__device__ __forceinline__ unsigned int fenc(float f) {
    unsigned int u = __float_as_uint(f);
    return (u & 0x80000000u) ? ~u : (u | 0x80000000u);
}
__device__ __forceinline__ float fdec(unsigned int e) {
    unsigned int u = (e & 0x80000000u) ? (e ^ 0x80000000u) : ~e;
    return __uint_as_float(u);
}

__device__ __forceinline__ void decode_tile(int tile, int& b, int& y, int& x0) {
    x0 = (tile & 7) * 16;          // W/16 = 8 tiles
    y = (tile >> 3) & (HH - 1);
    b = tile >> 10;
}

__global__ void k_init_minmax(unsigned int* mm) {
    mm[0] = 0xFFFFFFFFu;  // encoded running min
    mm[1] = 0u;           // encoded running max
}

// ---------------- Stage 1: 3x3 conv 32->192, pad 1 (implicit GEMM, K=288) ----------------
__global__ __launch_bounds__(256)
void k_input_conv(const float* __restrict__ x, const float* __restrict__ w_in,
                  const float* __restrict__ b_in, float* __restrict__ ic) {
    const int lane = threadIdx.x & 31;
    const int tile = blockIdx.x * 8 + (threadIdx.x >> 5);
    int b, y, x0; decode_tile(tile, b, y, x0);
    const int m = lane & 15;
    const int hi = lane >> 4;
    const int xx = x0 + m;

    v8f acc[12];
#pragma unroll
    for (int n = 0; n < 12; ++n) {
        float bias = b_in[n * 16 + m];
#pragma unroll
        for (int r = 0; r < 8; ++r) acc[n][r] = bias;
    }

    const float* xb = x + (size_t)b * CIN * HWSZ;

#pragma unroll 1
    for (int ks = 0; ks < 72; ++ks) {
        const int kbase = ks * 4 + hi * 2;
        v2f a;
#pragma unroll
        for (int j = 0; j < 2; ++j) {
            int k = kbase + j;
            int ci = k / 9;
            int tap = k - ci * 9;
            int ky = tap / 3, kx = tap - ky * 3;
            int yy = y + ky - 1;
            int xj = xx + kx - 1;
            bool inb = (yy >= 0) && (yy < HH) && (xj >= 0) && (xj < WW);
            int yyc = min(max(yy, 0), HH - 1);
            int xjc = min(max(xj, 0), WW - 1);
            float v = xb[(size_t)ci * HWSZ + yyc * WW + xjc];
            a[j] = inb ? v : 0.0f;
        }
#pragma unroll
        for (int n = 0; n < 12; ++n) {
            const float* wr = w_in + (size_t)(n * 16 + m) * 288 + kbase;
            v2f bf; bf[0] = wr[0]; bf[1] = wr[1];
            acc[n] = __builtin_amdgcn_wmma_f32_16x16x4_f32(
                false, a, false, bf, (short)0, acc[n], false, false);
        }
    }

    float* icb = ic + (size_t)b * C3 * HWSZ + (size_t)y * WW + x0 + hi * 8;
#pragma unroll
    for (int n = 0; n < 12; ++n) {
        float* p = icb + (size_t)(n * 16 + m) * HWSZ;
        f4 lo = {acc[n][0], acc[n][1], acc[n][2], acc[n][3]};
        f4 hi4 = {acc[n][4], acc[n][5], acc[n][6], acc[n][7]};
        *(f4*)p = lo;
        *(f4*)(p + 4) = hi4;
    }
}

// ---------------- Stage 2: 5x5 conv 96->16, pad 2 (implicit GEMM, K=2400) + min/max ----------------
__global__ __launch_bounds__(256)
void k_sub_conv(const float* __restrict__ x, const float* __restrict__ h_prev,
                const float* __restrict__ w_sub, const float* __restrict__ b_sub,
                float* __restrict__ cc, unsigned int* __restrict__ mm) {
    const int lane = threadIdx.x & 31;
    const int tile = blockIdx.x * 8 + (threadIdx.x >> 5);
    int b, y, x0; decode_tile(tile, b, y, x0);
    const int m = lane & 15;
    const int hi = lane >> 4;
    const int xx = x0 + m;

    v8f acc;
    {
        float bias = b_sub[m];
#pragma unroll
        for (int r = 0; r < 8; ++r) acc[r] = bias;
    }

    const float* xb = x + (size_t)b * CIN * HWSZ;
    const float* hb = h_prev + (size_t)b * HID * HWSZ;

#pragma unroll 1
    for (int ks = 0; ks < 600; ++ks) {
        const int kbase = ks * 4 + hi * 2;
        v2f a;
#pragma unroll
        for (int j = 0; j < 2; ++j) {
            int k = kbase + j;
            int ci = k / 25;
            int tap = k - ci * 25;
            int ky = tap / 5, kx = tap - ky * 5;
            int yy = y + ky - 2;
            int xj = xx + kx - 2;
            bool inb = (yy >= 0) && (yy < HH) && (xj >= 0) && (xj < WW);
            int yyc = min(max(yy, 0), HH - 1);
            int xjc = min(max(xj, 0), WW - 1);
            const float* src = (ci < CIN) ? (xb + (size_t)ci * HWSZ)
                                          : (hb + (size_t)(ci - CIN) * HWSZ);
            float v = src[yyc * WW + xjc];
            a[j] = inb ? v : 0.0f;
        }
        v2f bf;
        const float* wr = w_sub + (size_t)m * 2400 + kbase;
        bf[0] = wr[0]; bf[1] = wr[1];
        acc = __builtin_amdgcn_wmma_f32_16x16x4_f32(
            false, a, false, bf, (short)0, acc, false, false);
    }

    float* p = cc + ((size_t)b * CCCH + m) * HWSZ + (size_t)y * WW + x0 + hi * 8;
    f4 lo = {acc[0], acc[1], acc[2], acc[3]};
    f4 hi4 = {acc[4], acc[5], acc[6], acc[7]};
    *(f4*)p = lo;
    *(f4*)(p + 4) = hi4;

    float mn = acc[0], mx = acc[0];
#pragma unroll
    for (int r = 1; r < 8; ++r) { mn = fminf(mn, acc[r]); mx = fmaxf(mx, acc[r]); }
#pragma unroll
    for (int off = 16; off > 0; off >>= 1) {
        mn = fminf(mn, __shfl_xor(mn, off, 32));
        mx = fmaxf(mx, __shfl_xor(mx, off, 32));
    }
    if (lane == 0) {
        atomicMin(&mm[0], fenc(mn));
        atomicMax(&mm[1], fenc(mx));
    }
}

// ---------------- Stage 3: fused 4x(grid-sample + 64->192 proj) + GRU gating ----------------
__global__ __launch_bounds__(256)
void k_warp_proj_gate(const float* __restrict__ h_prev, const float* __restrict__ w_proj,
                      const float* __restrict__ b_proj, const float* __restrict__ ic,
                      const float* __restrict__ cc, const unsigned int* __restrict__ mm,
                      float* __restrict__ out) {
    const int lane = threadIdx.x & 31;
    const int tile = blockIdx.x * 8 + (threadIdx.x >> 5);
    int b, y, x0; decode_tile(tile, b, y, x0);
    const int m = lane & 15;
    const int hi = lane >> 4;
    const int xx = x0 + m;

    const float mn = fdec(mm[0]);
    const float mx = fdec(mm[1]);
    const float inv = 1.0f / (mx - mn);

    // per-pixel flow grids (channel 2i = x coord, 2i+1 = y coord)
    float gx[NW], gy[NW];
    const float* ccb = cc + (size_t)b * CCCH * HWSZ + (size_t)y * WW + xx;
#pragma unroll
    for (int i = 0; i < NW; ++i) {
        float tx = (ccb[(size_t)(2 * i) * HWSZ] - mn) * inv;
        float ty = (ccb[(size_t)(2 * i + 1) * HWSZ] - mn) * inv;
        gx[i] = tx * (float)WW - 0.5f;   // ((2t-1)+1)*W/2 - 0.5
        gy[i] = ty * (float)HH - 0.5f;
    }

    v8f acc[12];
#pragma unroll
    for (int n = 0; n < 12; ++n) {
        int o = n * 16 + m;
        float bias = b_proj[o] + b_proj[C3 + o] + b_proj[2 * C3 + o] + b_proj[3 * C3 + o];
#pragma unroll
        for (int r = 0; r < 8; ++r) acc[n][r] = bias;
    }

    const float* hb = h_prev + (size_t)b * HID * HWSZ;

#pragma unroll 1
    for (int i = 0; i < NW; ++i) {
        float fx = floorf(gx[i]), fy = floorf(gy[i]);
        float wx1 = gx[i] - fx, wy1 = gy[i] - fy;
        float wx0 = 1.0f - wx1, wy0 = 1.0f - wy1;
        int x0i = (int)fx, y0i = (int)fy;
        int x1i = x0i + 1, y1i = y0i + 1;
        float vx0 = (x0i >= 0 && x0i < WW) ? 1.0f : 0.0f;
        float vx1 = (x1i >= 0 && x1i < WW) ? 1.0f : 0.0f;
        float vy0 = (y0i >= 0 && y0i < HH) ? 1.0f : 0.0f;
        float vy1 = (y1i >= 0 && y1i < HH) ? 1.0f : 0.0f;
        float w00 = wy0 * wx0 * vy0 * vx0;
        float w01 = wy0 * wx1 * vy0 * vx1;
        float w10 = wy1 * wx0 * vy1 * vx0;
        float w11 = wy1 * wx1 * vy1 * vx1;
        int xc0 = min(max(x0i, 0), WW - 1), xc1 = min(max(x1i, 0), WW - 1);
        int yc0 = min(max(y0i, 0), HH - 1), yc1 = min(max(y1i, 0), HH - 1);
        int i00 = yc0 * WW + xc0, i01 = yc0 * WW + xc1;
        int i10 = yc1 * WW + xc0, i11 = yc1 * WW + xc1;

        const float* wp = w_proj + (size_t)i * C3 * HID;
#pragma unroll 1
        for (int ks = 0; ks < 16; ++ks) {
            const int kbase = ks * 4 + hi * 2;
            v2f a;
#pragma unroll
            for (int j = 0; j < 2; ++j) {
                const float* hc = hb + (size_t)(kbase + j) * HWSZ;
                a[j] = w00 * hc[i00] + w01 * hc[i01] + w10 * hc[i10] + w11 * hc[i11];
            }
#pragma unroll
            for (int n = 0; n < 12; ++n) {
                const float* wr = wp + (size_t)(n * 16 + m) * HID + kbase;
                v2f bf; bf[0] = wr[0]; bf[1] = wr[1];
                acc[n] = __builtin_amdgcn_wmma_f32_16x16x4_f32(
                    false, a, false, bf, (short)0, acc[n], false, false);
            }
        }
    }

    // GRU gating: traj ch [0,64)=h_z, [64,128)=h_r, [128,192)=h_h
#pragma unroll
    for (int cg = 0; cg < 4; ++cg) {
        int c = cg * 16 + m;
        const size_t pix = (size_t)y * WW + x0 + hi * 8;
        const float* pxz = ic + ((size_t)b * C3 + c) * HWSZ + pix;
        const float* pxr = pxz + (size_t)HID * HWSZ;
        const float* pxh = pxz + (size_t)(2 * HID) * HWSZ;
        const float* ph = h_prev + ((size_t)b * HID + c) * HWSZ + pix;
        float* po = out + ((size_t)b * HID + c) * HWSZ + pix;
        v8f hz = acc[cg], hr = acc[cg + 4], hh = acc[cg + 8];
        f4 o0, o1;
#pragma unroll
        for (int r = 0; r < 8; ++r) {
            float z = 1.0f / (1.0f + __expf(-(pxz[r] + hz[r])));
            float rr = 1.0f / (1.0f + __expf(-(pxr[r] + hr[r])));
            float hv = pxh[r] + rr * hh[r];
            hv = hv >= 0.0f ? hv : 0.2f * hv;
            float res = (1.0f - z) * hv + z * ph[r];
            if (r < 4) o0[r] = res; else o1[r - 4] = res;
        }
        *(f4*)po = o0;
        *(f4*)(po + 4) = o1;
    }
}

extern "C" void kernel_launch(void* const* d_in, const int* in_sizes, int n_in,
                              void* d_out, int out_size, void* d_ws, size_t ws_size,
                              hipStream_t stream) {
    const float* x      = (const float*)d_in[0];
    const float* h_prev = (const float*)d_in[1];
    const float* w_in   = (const float*)d_in[2];
    const float* b_in   = (const float*)d_in[3];
    const float* w_proj = (const float*)d_in[4];
    const float* b_proj = (const float*)d_in[5];
    const float* w_sub  = (const float*)d_in[6];
    const float* b_sub  = (const float*)d_in[7];
    float* out = (float*)d_out;

    float* ws = (float*)d_ws;
    float* ic = ws;                                            // 16*192*128*128 floats
    float* cc = ic + (size_t)BATCH * C3 * HWSZ;                // 16*16*128*128 floats
    unsigned int* mm = (unsigned int*)(cc + (size_t)BATCH * CCCH * HWSZ);

    const int tiles = BATCH * HH * (WW / 16);   // 16384
    const int blocks = tiles / 8;               // 8 waves per 256-thread block

    k_init_minmax<<<1, 1, 0, stream>>>(mm);
    k_input_conv<<<blocks, 256, 0, stream>>>(x, w_in, b_in, ic);
    k_sub_conv<<<blocks, 256, 0, stream>>>(x, h_prev, w_sub, b_sub, cc, mm);
    k_warp_proj_gate<<<blocks, 256, 0, stream>>>(h_prev, w_proj, b_proj, ic, cc, mm, out);
}